// RGCNLayer_26998164423429
// MI455X (gfx1250) — compile-verified
//
#include <hip/hip_runtime.h>
#include <hip/hip_bf16.h>

#define N_NODES 50000
#define N_EDGES 1600000
#define FEAT    128

typedef float v2f __attribute__((ext_vector_type(2)));
typedef float v8f __attribute__((ext_vector_type(8)));

// ---------------------------------------------------------------------------
// K0: out[n,f] = bias[f];  m_ord = 0 (ordinal -inf);  den = 0
// ---------------------------------------------------------------------------
__global__ void k_init(float* __restrict__ out, const float* __restrict__ bias,
                       unsigned* __restrict__ m_ord, float* __restrict__ den) {
    int i = blockIdx.x * blockDim.x + threadIdx.x;
    if (i < N_NODES * FEAT) out[i] = bias[i & (FEAT - 1)];
    if (i < N_NODES) { m_ord[i] = 0u; den[i] = 0.0f; }
}

// ---------------------------------------------------------------------------
// K1: z = h @ W^T via V_WMMA_F32_16X16X4_F32.
// One wave computes one 16x16 tile of z. Block = 256 threads = 8 waves;
// wave w handles column-tile w (128/16 = 8 tiles), blockIdx.x is the row tile
// (50000/16 = 3125 tiles, exact).
//
// ISA 32-bit layouts (wave32):
//   A (16x4, MxK):  lanes 0-15 hold M=lane; VGPR0/1 = K=0/1 (lo half) or
//                   K=2/3 (hi half, lanes 16-31).
//   B (4x16, KxN):  VGPR v, lanes 0-15 = row K=v, lanes 16-31 = row K=v+2;
//                   N = lane%16.   B[k][n] = W[n][k]  (z = h @ W^T).
//   C/D (16x16):    VGPR v, lanes 0-15 -> M=v, lanes 16-31 -> M=v+8; N=lane%16.
// ---------------------------------------------------------------------------
__global__ void k_gemm_z(const float* __restrict__ h, const float* __restrict__ W,
                         float* __restrict__ z) {
    const int wave = threadIdx.x >> 5;      // column tile 0..7
    const int lane = threadIdx.x & 31;
    const int half = lane >> 4;             // 0: K pair {0,1}, 1: K pair {2,3}
    const int l16  = lane & 15;
    const int rowBase = blockIdx.x * 16;
    const int colBase = wave * 16;

    const float* __restrict__ arow = h + (size_t)(rowBase + l16) * FEAT + 2 * half;
    const float* __restrict__ brow = W + (size_t)(colBase + l16) * FEAT + 2 * half;

    v8f c = {};
    #pragma unroll
    for (int kk = 0; kk < FEAT / 4; ++kk) {
        v2f a, b;
        a.x = arow[kk * 4 + 0];
        a.y = arow[kk * 4 + 1];
        b.x = brow[kk * 4 + 0];
        b.y = brow[kk * 4 + 1];
        c = __builtin_amdgcn_wmma_f32_16x16x4_f32(
                /*neg_a=*/false, a, /*neg_b=*/false, b,
                /*c_mod=*/(short)0, c, /*reuse_a=*/false, /*reuse_b=*/false);
    }

    float* __restrict__ zcol = z + colBase + l16;
    #pragma unroll
    for (int v = 0; v < 8; ++v) {
        int row = rowBase + v + 8 * half;
        zcol[(size_t)row * FEAT] = c[v];
    }
}

// ---------------------------------------------------------------------------
// K2: s_src[n] = z[n,:] . w1 ; s_dst[n] = z[n,:] . w2  (wave per node)
// ---------------------------------------------------------------------------
__global__ void k_scores(const float* __restrict__ z, const float* __restrict__ attn_w,
                         float* __restrict__ s_src, float* __restrict__ s_dst) {
    const int node = blockIdx.x * 8 + (threadIdx.x >> 5);
    const int lane = threadIdx.x & 31;
    if (node >= N_NODES) return;

    float4 zv = ((const float4*)(z + (size_t)node * FEAT))[lane];
    float4 a1 = ((const float4*)attn_w)[lane];
    float4 a2 = ((const float4*)(attn_w + FEAT))[lane];

    float d1 = zv.x * a1.x + zv.y * a1.y + zv.z * a1.z + zv.w * a1.w;
    float d2 = zv.x * a2.x + zv.y * a2.y + zv.z * a2.z + zv.w * a2.w;
    #pragma unroll
    for (int off = 16; off > 0; off >>= 1) {
        d1 += __shfl_down(d1, off, 32);
        d2 += __shfl_down(d2, off, 32);
    }
    if (lane == 0) { s_src[node] = d1; s_dst[node] = d2; }
}

// Order-preserving float <-> uint mapping for atomic max on signed floats.
__device__ __forceinline__ unsigned f2ord(float f) {
    unsigned u = __float_as_uint(f);
    return (u >> 31) ? ~u : (u | 0x80000000u);
}
__device__ __forceinline__ float ord2f(unsigned o) {
    return (o >> 31) ? __uint_as_float(o & 0x7FFFFFFFu) : __uint_as_float(~o);
}

// ---------------------------------------------------------------------------
// K3: e = leaky_relu(s_src[src]+s_dst[dst]); segment max via ordinal atomicMax
// ---------------------------------------------------------------------------
__global__ void k_edge_max(const float* __restrict__ s_src, const float* __restrict__ s_dst,
                           const int* __restrict__ src, const int* __restrict__ dst,
                           float* __restrict__ e, unsigned* __restrict__ m_ord) {
    int i = blockIdx.x * blockDim.x + threadIdx.x;
    if (i >= N_EDGES) return;
    float v = s_src[src[i]] + s_dst[dst[i]];
    v = (v > 0.0f) ? v : 0.01f * v;
    e[i] = v;
    atomicMax(&m_ord[dst[i]], f2ord(v));
}

// ---------------------------------------------------------------------------
// K4: num = exp(e - m[dst]); den[dst] += num  (num overwrites e)
// ---------------------------------------------------------------------------
__global__ void k_edge_sum(const int* __restrict__ dst, const unsigned* __restrict__ m_ord,
                           float* __restrict__ e, float* __restrict__ den) {
    int i = blockIdx.x * blockDim.x + threadIdx.x;
    if (i >= N_EDGES) return;
    int d = dst[i];
    float num = __expf(e[i] - ord2f(m_ord[d]));
    e[i] = num;
    atomicAdd(&den[d], num);
}

// ---------------------------------------------------------------------------
// K5: out[dst,:] += rel_emb[etype] * (num/den[dst]) * z[src,:]
// One wave per edge: lane moves one float4 (512B coalesced per wave);
// gathers hit L2 (z fits in 192MB L2), atomics resolve at L2.
// ---------------------------------------------------------------------------
__global__ void k_edge_scatter(const float* __restrict__ z, const float* __restrict__ e,
                               const float* __restrict__ den, const float* __restrict__ rel_emb,
                               const int* __restrict__ src, const int* __restrict__ dst,
                               const int* __restrict__ etype, float* __restrict__ out) {
    const int edge = blockIdx.x * 8 + (threadIdx.x >> 5);
    const int lane = threadIdx.x & 31;
    if (edge >= N_EDGES) return;

    int s = src[edge], d = dst[edge];
    float coef = rel_emb[etype[edge]] * e[edge] / den[d];

    float4 zv = ((const float4*)(z + (size_t)s * FEAT))[lane];
    float* o = out + (size_t)d * FEAT + lane * 4;
    atomicAdd(o + 0, coef * zv.x);
    atomicAdd(o + 1, coef * zv.y);
    atomicAdd(o + 2, coef * zv.z);
    atomicAdd(o + 3, coef * zv.w);
}

// ---------------------------------------------------------------------------
extern "C" void kernel_launch(void* const* d_in, const int* in_sizes, int n_in,
                              void* d_out, int out_size, void* d_ws, size_t ws_size,
                              hipStream_t stream) {
    const float* h       = (const float*)d_in[0];
    const float* W       = (const float*)d_in[1];
    const float* attn_w  = (const float*)d_in[2];
    const float* rel_emb = (const float*)d_in[3];
    const float* bias_w  = (const float*)d_in[4];
    const int*   src     = (const int*)d_in[5];
    const int*   dst     = (const int*)d_in[6];
    const int*   etype   = (const int*)d_in[7];
    float* out = (float*)d_out;

    // Workspace layout (floats): z | s_src | s_dst | m_ord | den | e
    float*    z     = (float*)d_ws;
    float*    s_src = z + (size_t)N_NODES * FEAT;
    float*    s_dst = s_src + N_NODES;
    unsigned* m_ord = (unsigned*)(s_dst + N_NODES);
    float*    den   = (float*)(m_ord + N_NODES);
    float*    e     = den + N_NODES;

    // K0: init output with bias, reset reductions
    k_init<<<(N_NODES * FEAT + 255) / 256, 256, 0, stream>>>(out, bias_w, m_ord, den);

    // K1: WMMA GEMM  z = h @ W^T   (3125 row tiles x 8 col tiles; 8 waves/block)
    k_gemm_z<<<N_NODES / 16, 256, 0, stream>>>(h, W, z);

    // K2: attention scores per node (8 waves/block)
    k_scores<<<N_NODES / 8, 256, 0, stream>>>(z, attn_w, s_src, s_dst);

    // K3/K4: edge softmax reductions
    k_edge_max<<<(N_EDGES + 255) / 256, 256, 0, stream>>>(s_src, s_dst, src, dst, e, m_ord);
    k_edge_sum<<<(N_EDGES + 255) / 256, 256, 0, stream>>>(dst, m_ord, e, den);

    // K5: weighted gather/scatter (wave per edge)
    k_edge_scatter<<<N_EDGES / 8, 256, 0, stream>>>(z, e, den, rel_emb, src, dst, etype, out);
}